// FocalLoss_79182017069846
// MI455X (gfx1250) — compile-verified
//
#include <hip/hip_runtime.h>
#include <cstdint>
#include <cstddef>

// Problem constants (from reference)
#define A_N 200000
#define B_N 8
#define M_N 32
#define C_N 4

#define THREADS 256
#define NBLK ((A_N + THREADS - 1) / THREADS)   // 782

typedef unsigned int u32x4 __attribute__((ext_vector_type(4)));
typedef int          i32x8 __attribute__((ext_vector_type(8)));
typedef int          i32x4 __attribute__((ext_vector_type(4)));
typedef float        f32x4 __attribute__((ext_vector_type(4)));

// ---------------------------------------------------------------------------
// Main kernel: one anchor per thread. Annotations (8*32*5 f32 = 5120 B) are
// DMA'd into LDS by the Tensor Data Mover (one tensor_load_to_lds issued by
// wave 0), then each thread computes argmax-IoU over the 32 boxes per image,
// the focal term (only class channel 0 survives the target mask), and the
// smooth-L1 term for positives. All streamed global data is loaded with
// non-temporal hints (read-once). Per-block partials go to the workspace for
// a deterministic second pass.
// ---------------------------------------------------------------------------
__global__ __launch_bounds__(THREADS) void focal_main(
    const float* __restrict__ cls,   // (B, A, 4)
    const float* __restrict__ reg,   // (B, A, 4)
    const float* __restrict__ anc,   // (1, A, 4)
    const float* __restrict__ ann,   // (B, M, 5)
    float* __restrict__ part)        // (NBLK, 24)
{
    __shared__ float s_ann[B_N * M_N * 5];   // 5120 bytes
    __shared__ float s_red[THREADS / 32][24];

    const int tid  = threadIdx.x;
    const int wave = tid >> 5;
    const int lane = tid & 31;

    // ---- TDM: DMA annotations into LDS (wave 0 only; TDM ignores EXEC, so
    // gate with a wave-uniform branch). Tracked by TENSORcnt. ----
    if (wave == 0) {
        const uint64_t gaddr = (uint64_t)(uintptr_t)ann;
        const uint32_t laddr = (uint32_t)(uintptr_t)(&s_ann[0]);
        const uint32_t n     = (uint32_t)(B_N * M_N * 5);   // 1280 dwords

        u32x4 g0;
        g0[0] = 1u;                                          // count=1, user mode
        g0[1] = laddr;                                       // lds_addr (bytes)
        g0[2] = (uint32_t)gaddr;                             // global_addr[31:0]
        g0[3] = (uint32_t)((gaddr >> 32) & 0x1FFFFFFu)       // global_addr[56:32]
              | (2u << 30);                                  // type = 2 ("image")

        i32x8 g1;
        g1[0] = (int)(2u << 16);                 // workgroup_mask=0, data_size=4B
        g1[1] = (int)(n << 16);                  // tensor_dim0[15:0]
        g1[2] = (int)((n >> 16) | (1u << 16));   // tensor_dim0[31:16] | tensor_dim1=1
        g1[3] = (int)(n << 16);                  // tensor_dim1 hi=0 | tile_dim0=n
        g1[4] = 1;                               // tile_dim1=1, tile_dim2=0
        g1[5] = (int)n;                          // tensor_dim0_stride[31:0]
        g1[6] = 0;
        g1[7] = 0;

        const i32x4 gz4 = {0, 0, 0, 0};          // groups 2/3 unused (<=2D)
        const i32x8 gz8 = {0, 0, 0, 0, 0, 0, 0, 0};
        __builtin_amdgcn_tensor_load_to_lds(g0, g1, gz4, gz4, gz8, 0);
        __builtin_amdgcn_s_wait_tensorcnt(0);
    }
    __syncthreads();

    float cls_acc[B_N], pos_acc[B_N], reg_acc[B_N];
#pragma unroll
    for (int b = 0; b < B_N; ++b) { cls_acc[b] = 0.f; pos_acc[b] = 0.f; reg_acc[b] = 0.f; }

    const int a = blockIdx.x * THREADS + tid;
    if (a < A_N) {
        // Streamed once -> non-temporal load (TH_NT)
        const f32x4 an = __builtin_nontemporal_load((const f32x4*)anc + a);
        const float aw = an.z - an.x, ah = an.w - an.y;
        const float acx = an.x + 0.5f * aw, acy = an.y + 0.5f * ah;
        const float area_a = aw * ah;

#pragma unroll 1
        for (int b = 0; b < B_N; ++b) {
            const float* bx = &s_ann[b * M_N * 5];
            float best = -1.0f;
            int   bi   = 0;
#pragma unroll
            for (int m = 0; m < M_N; ++m) {
                const float x1 = bx[m * 5 + 0], y1 = bx[m * 5 + 1];
                const float x2 = bx[m * 5 + 2], y2 = bx[m * 5 + 3];
                float iw = fminf(an.z, x2) - fmaxf(an.x, x1);
                float ih = fminf(an.w, y2) - fmaxf(an.y, y1);
                iw = fmaxf(iw, 0.f); ih = fmaxf(ih, 0.f);
                const float inter = iw * ih;
                const float ua = fmaxf(area_a + (x2 - x1) * (y2 - y1) - inter, 1e-8f);
                const float iou = inter / ua;
                if (iou > best) { best = iou; bi = m; }   // first-max == argmax
            }
            const bool pos = (best >= 0.5f);
            const bool neg = (best <  0.4f);

            // Focal term: only class channel 0 survives the target mask.
            float p = __builtin_nontemporal_load(&cls[((size_t)b * A_N + a) * C_N]);
            p = fminf(fmaxf(p, 1e-5f), 1.f - 1e-5f);
            float l = 0.f;
            if (pos)      l = 0.25f * (1.f - p) * (1.f - p) * (-logf(p));
            else if (neg) l = 0.75f * p * p * (-logf(1.f - p));
            cls_acc[b] += l;

            if (pos) {
                pos_acc[b] += 1.f;
                const float* g = &s_ann[(b * M_N + bi) * 5];
                const float gw_raw = g[2] - g[0], gh_raw = g[3] - g[1];
                const float gcx = g[0] + 0.5f * gw_raw, gcy = g[1] + 0.5f * gh_raw;
                const float gw = fmaxf(gw_raw, 1.f), gh = fmaxf(gh_raw, 1.f);
                // Only positives touch regressions: EXEC-masked lanes skip
                // whole cachelines, cutting real HBM traffic.
                const f32x4 rg =
                    __builtin_nontemporal_load((const f32x4*)reg + (size_t)b * A_N + a);
                const float rt0 = (gcx - acx) / aw / 0.1f;
                const float rt1 = (gcy - acy) / ah / 0.1f;
                const float rt2 = logf(gw / aw) / 0.2f;
                const float rt3 = logf(gh / ah) / 0.2f;
                float s = 0.f, d;
                d = fabsf(rt0 - rg.x); s += (d <= (1.f / 9.f)) ? 4.5f * d * d : d - (0.5f / 9.f);
                d = fabsf(rt1 - rg.y); s += (d <= (1.f / 9.f)) ? 4.5f * d * d : d - (0.5f / 9.f);
                d = fabsf(rt2 - rg.z); s += (d <= (1.f / 9.f)) ? 4.5f * d * d : d - (0.5f / 9.f);
                d = fabsf(rt3 - rg.w); s += (d <= (1.f / 9.f)) ? 4.5f * d * d : d - (0.5f / 9.f);
                reg_acc[b] += s;
            }
        }
    }

    // ---- wave32 shuffle reduction (offsets 16..1), then cross-wave via LDS ----
#pragma unroll
    for (int b = 0; b < B_N; ++b) {
#pragma unroll
        for (int off = 16; off > 0; off >>= 1) {
            cls_acc[b] += __shfl_down(cls_acc[b], off);
            pos_acc[b] += __shfl_down(pos_acc[b], off);
            reg_acc[b] += __shfl_down(reg_acc[b], off);
        }
    }
    if (lane == 0) {
#pragma unroll
        for (int b = 0; b < B_N; ++b) {
            s_red[wave][b * 3 + 0] = cls_acc[b];
            s_red[wave][b * 3 + 1] = pos_acc[b];
            s_red[wave][b * 3 + 2] = reg_acc[b];
        }
    }
    __syncthreads();
    if (tid < 24) {
        float s = 0.f;
#pragma unroll
        for (int w = 0; w < THREADS / 32; ++w) s += s_red[w][tid];
        part[(size_t)blockIdx.x * 24 + tid] = s;
    }
}

// ---------------------------------------------------------------------------
// Finalize: deterministic fixed-order sum over block partials, then per-image
// normalization and batch mean. Tiny; one workgroup.
// ---------------------------------------------------------------------------
__global__ __launch_bounds__(64) void focal_final(
    const float* __restrict__ part, float* __restrict__ out)
{
    __shared__ float s[24];
    const int t = threadIdx.x;
    if (t < 24) {
        float acc = 0.f;
        for (int i = 0; i < NBLK; ++i) acc += part[(size_t)i * 24 + t];
        s[t] = acc;
    }
    __syncthreads();
    if (t == 0) {
        float cm = 0.f, rm = 0.f;
#pragma unroll
        for (int b = 0; b < B_N; ++b) {
            const float cs = s[b * 3 + 0];
            const float np = s[b * 3 + 1];
            const float rs = s[b * 3 + 2];
            const float den = fmaxf(np, 1.f);
            cm += cs / den;
            rm += (np > 0.f) ? rs / (4.f * den) : 0.f;
        }
        out[0] = cm / (float)B_N;
        out[1] = rm / (float)B_N;
    }
}

extern "C" void kernel_launch(void* const* d_in, const int* in_sizes, int n_in,
                              void* d_out, int out_size, void* d_ws, size_t ws_size,
                              hipStream_t stream) {
    const float* cls = (const float*)d_in[0];   // (B, A, 4)
    const float* reg = (const float*)d_in[1];   // (B, A, 4)
    const float* anc = (const float*)d_in[2];   // (1, A, 4)
    const float* ann = (const float*)d_in[3];   // (B, M, 5)
    float* part = (float*)d_ws;                 // NBLK*24 floats = ~75 KB

    focal_main<<<NBLK, THREADS, 0, stream>>>(cls, reg, anc, ann, part);
    focal_final<<<1, 64, 0, stream>>>(part, (float*)d_out);
}